// LinearAttention_64467459113130
// MI455X (gfx1250) — compile-verified
//
#include <hip/hip_runtime.h>

// ---------------------------------------------------------------------------
// Linear attention pipeline for MI455X (gfx1250), fp32 throughout using
// V_WMMA_F32_16X16X4_F32 (wave32, 16x16 C/D in 8 VGPRs, A/B in 2 VGPRs).
//
// Sizes: B=8, C=128, HEADS=4, DH=64, N=128*128=16384, INNER=256, QKV rows=768.
// N is compile-time (16384) so all row strides become VMEM immediate offsets;
// k-loops are software-pipelined (prefetch distance 1) so loads for step k+1
// are in flight while step k's WMMAs execute.
// ---------------------------------------------------------------------------

#define NTOK 16384

typedef float v2f __attribute__((ext_vector_type(2)));
typedef float v8f __attribute__((ext_vector_type(8)));

static __device__ __forceinline__ v8f wmma4(v2f a, v2f b, v8f c) {
  // D = A(16x4,f32) * B(4x16,f32) + C(16x16,f32)
  return __builtin_amdgcn_wmma_f32_16x16x4_f32(
      /*neg_a=*/false, a, /*neg_b=*/false, b,
      /*c_mod=*/(short)0, c, /*reuse_a=*/false, /*reuse_b=*/false);
}

static __device__ __forceinline__ v8f v8f_zero() {
  v8f z = {0.f, 0.f, 0.f, 0.f, 0.f, 0.f, 0.f, 0.f};
  return z;
}

// ---------------------------------------------------------------------------
// Generic fp32 WMMA GEMM:  Out[z][o][n] = sum_c W[z][o][c] * X[z][c][n] (+bias[o])
// Per-wave tile: 16(M) x 64(N); block = 8 waves -> 32(M) x 256(N).
// z decomposed as z_hi = z / zdiv, z_lo = z % zdiv with independent strides.
// C (the K dim) is a template parameter: 128 (qkv), 64 (attn), 256 (final).
// All launch dims divide exactly: no bounds checks, EXEC stays all-ones.
// ---------------------------------------------------------------------------
template <int C>
__global__ __launch_bounds__(256) void gemm_wmma_f32(
    const float* __restrict__ W, const float* __restrict__ X,
    float* __restrict__ Out, const float* __restrict__ bias,
    int zdiv, long sW1, long sW0, long sX1, long sX0, long sO1, long sO0)
{
  const int tid  = threadIdx.x;
  const int wave = tid >> 5;
  const int lane = tid & 31;
  const int half = lane >> 4;   // lane half selects K pair (A/B) and M half (C/D)
  const int l16  = lane & 15;

  const int m0 = blockIdx.x * 32 + (wave & 1) * 16;
  const int n0 = blockIdx.y * 256 + (wave >> 1) * 64;

  const long z  = blockIdx.z;
  const long zh = z / zdiv;
  const long zl = z % zdiv;
  // Marching pointers; all intra-step offsets are immediates.
  const float* wp = W + zh * sW1 + zl * sW0 + (long)(m0 + l16) * C + 2 * half;
  const float* xp = X + zh * sX1 + zl * sX0 + (long)(2 * half) * NTOK + n0 + l16;
  float*       Op = Out + zh * sO1 + zl * sO0;

  v8f acc[4] = {v8f_zero(), v8f_zero(), v8f_zero(), v8f_zero()};

  // Stage 0 fragments.
  v2f a = *(const v2f*)wp;
  v2f bf[4];
#pragma unroll
  for (int t = 0; t < 4; ++t) { bf[t].x = xp[t * 16]; bf[t].y = xp[t * 16 + NTOK]; }

  // Pipelined main loop: issue k+1 loads, then k WMMAs.
  for (int k0 = 0; k0 < C - 4; k0 += 4) {
    wp += 4;
    xp += (long)4 * NTOK;
    __builtin_prefetch(xp + (long)16 * NTOK, 0, 3);   // ~4 steps ahead
    v2f a_n = *(const v2f*)wp;
    v2f bf_n[4];
#pragma unroll
    for (int t = 0; t < 4; ++t) { bf_n[t].x = xp[t * 16]; bf_n[t].y = xp[t * 16 + NTOK]; }
#pragma unroll
    for (int t = 0; t < 4; ++t) acc[t] = wmma4(a, bf[t], acc[t]);
    a = a_n;
#pragma unroll
    for (int t = 0; t < 4; ++t) bf[t] = bf_n[t];
  }
#pragma unroll
  for (int t = 0; t < 4; ++t) acc[t] = wmma4(a, bf[t], acc[t]);

#pragma unroll
  for (int t = 0; t < 4; ++t) {
#pragma unroll
    for (int v = 0; v < 8; ++v) {
      const int m = m0 + v + 8 * half;            // C/D: M split across halves
      float val = acc[t][v];
      if (bias) val += bias[m];
      Op[(long)m * NTOK + n0 + t * 16 + l16] = val;
    }
  }
}

// ---------------------------------------------------------------------------
// Row-wise softmax stats for k: per row (b, h*64+d) of 16384 values,
// compute max and 1/sum(exp(x-max)).  One 256-thread block per row.
// ---------------------------------------------------------------------------
__global__ __launch_bounds__(256) void k_rowstats(
    const float* __restrict__ qkv, float* __restrict__ kmax, float* __restrict__ kinv)
{
  __shared__ float red[256];
  const int r   = blockIdx.x;          // 0..2047
  const int b   = r >> 8;
  const int rem = r & 255;             // h*64 + d
  const float* row = qkv + (long)(b * 768 + 256 + rem) * NTOK;
  const int tid = threadIdx.x;

  float m = -3.4e38f;
  for (int i = tid; i < NTOK; i += 256) m = fmaxf(m, row[i]);
  red[tid] = m; __syncthreads();
  for (int s = 128; s > 0; s >>= 1) {
    if (tid < s) red[tid] = fmaxf(red[tid], red[tid + s]);
    __syncthreads();
  }
  m = red[0]; __syncthreads();

  float sum = 0.f;
  for (int i = tid; i < NTOK; i += 256) sum += __expf(row[i] - m);
  red[tid] = sum; __syncthreads();
  for (int s = 128; s > 0; s >>= 1) {
    if (tid < s) red[tid] += red[tid + s];
    __syncthreads();
  }
  if (tid == 0) { kmax[r] = m; kinv[r] = 1.0f / red[0]; }
}

// ---------------------------------------------------------------------------
// context^T partials: for (b,h), ctxT[e][d] = sum_n softmax_n(k)[d][n]*v[e][n].
// K-dim of the WMMA is n; softmax(k) fused into the A fragment.  Each wave:
// 16(d) x 64(e) over a 1024-wide n sub-chunk.  Pipelined like the GEMM.
// Partials per chunk (no atomics -> deterministic), reduced by ctx_reduce.
// grid = (8 chunks of 2048, 32 bh), block = 256 (4 d-tiles x 2 n-halves).
// ---------------------------------------------------------------------------
__global__ __launch_bounds__(256) void context_wmma(
    const float* __restrict__ qkv, const float* __restrict__ kmax,
    const float* __restrict__ kinv, float* __restrict__ part)
{
  const int tid  = threadIdx.x;
  const int wave = tid >> 5;
  const int lane = tid & 31;
  const int half = lane >> 4;
  const int l16  = lane & 15;

  const int dt = wave & 3;             // d tile 0..3
  const int ns = wave >> 2;            // n half 0..1
  const int bh = blockIdx.y;
  const int b  = bh >> 2;
  const int h  = bh & 3;
  const int d0 = dt * 16;
  const int chunk = blockIdx.x * 2 + ns;   // 0..15, each 1024 wide
  const long nbase = (long)chunk * 1024;

  const int drow = d0 + l16;
  const float km = kmax[bh * 64 + drow];
  const float ki = kinv[bh * 64 + drow];

  // Marching pointers; the per-t V-row offsets (t * 16 rows = t * 1 MiB) are
  // VMEM immediates.
  const float* kp = qkv + (long)(b * 768 + 256 + h * 64 + drow) * NTOK + nbase + 2 * half;
  const float* vp = qkv + (long)(b * 768 + 512 + h * 64 + l16) * NTOK + nbase + 2 * half;

  v8f acc[4] = {v8f_zero(), v8f_zero(), v8f_zero(), v8f_zero()};

  v2f kr = *(const v2f*)kp;
  v2f vf[4];
#pragma unroll
  for (int t = 0; t < 4; ++t) vf[t] = *(const v2f*)(vp + (long)t * 16 * NTOK);

  for (int nk = 0; nk < 1024 - 4; nk += 4) {
    kp += 4;
    vp += 4;
    v2f kr_n = *(const v2f*)kp;
    v2f vf_n[4];
#pragma unroll
    for (int t = 0; t < 4; ++t) vf_n[t] = *(const v2f*)(vp + (long)t * 16 * NTOK);
    v2f a;
    a.x = __expf(kr.x - km) * ki;
    a.y = __expf(kr.y - km) * ki;
#pragma unroll
    for (int t = 0; t < 4; ++t) acc[t] = wmma4(a, vf[t], acc[t]);
    kr = kr_n;
#pragma unroll
    for (int t = 0; t < 4; ++t) vf[t] = vf_n[t];
  }
  {
    v2f a;
    a.x = __expf(kr.x - km) * ki;
    a.y = __expf(kr.y - km) * ki;
#pragma unroll
    for (int t = 0; t < 4; ++t) acc[t] = wmma4(a, vf[t], acc[t]);
  }

  // part[((bh*16 + chunk)*64 + e)*64 + d]
  float* p = part + (long)(bh * 16 + chunk) * 4096;
#pragma unroll
  for (int t = 0; t < 4; ++t) {
#pragma unroll
    for (int v = 0; v < 8; ++v) {
      p[(t * 16 + l16) * 64 + d0 + v + 8 * half] = acc[t][v];
    }
  }
}

// Reduce 16 n-chunk partials -> ctxT[bh][e][d] (32 * 64 * 64 = 131072 floats)
__global__ __launch_bounds__(256) void ctx_reduce(
    const float* __restrict__ part, float* __restrict__ ctxT)
{
  const int idx = blockIdx.x * 256 + threadIdx.x;   // grid covers 131072
  const int bh = idx >> 12;
  const int ed = idx & 4095;
  const float* p = part + (long)bh * 65536 + ed;
  float s = 0.f;
#pragma unroll
  for (int c = 0; c < 16; ++c) s += p[c * 4096];
  ctxT[idx] = s;
}

// ---------------------------------------------------------------------------
// Column softmax of q over d (64 values) in place. One thread per (b,h,n)
// column; consecutive threads -> consecutive n -> coalesced per d-row.
// ---------------------------------------------------------------------------
__global__ __launch_bounds__(256) void q_softmax(float* __restrict__ qkv)
{
  const int idx = blockIdx.x * 256 + threadIdx.x;   // 524288 columns
  const int b   = idx >> 16;
  const int rem = idx & 65535;
  const int h   = rem >> 14;
  const int n   = rem & 16383;
  float* base = qkv + (long)(b * 768 + h * 64) * NTOK + n;

  float m = -3.4e38f;
  for (int d = 0; d < 64; ++d) m = fmaxf(m, base[(long)d * NTOK]);
  float s = 0.f;
  for (int d = 0; d < 64; ++d) s += __expf(base[(long)d * NTOK] - m);
  const float inv = 1.0f / s;
  for (int d = 0; d < 64; ++d) {
    const float x = base[(long)d * NTOK];
    base[(long)d * NTOK] = __expf(x - m) * inv;
  }
}

// ---------------------------------------------------------------------------
// Launch: inputs are x, w_qkv, w_out, b_out (all fp32).
// Workspace layout (floats):
//   qkv  : 8*768*16384       = 100663296   (q rows 0-255 / k 256-511 / v 512-767)
//   part : 32*16*64*64       =   2097152
//   ctxT : 32*64*64          =    131072
//   kmax : 2048, kinv : 2048
// attn output is written into the (dead) k region of qkv before the final GEMM.
// ---------------------------------------------------------------------------
extern "C" void kernel_launch(void* const* d_in, const int* in_sizes, int n_in,
                              void* d_out, int out_size, void* d_ws, size_t ws_size,
                              hipStream_t stream) {
  (void)in_sizes; (void)n_in; (void)out_size; (void)ws_size;
  const float* x     = (const float*)d_in[0];
  const float* w_qkv = (const float*)d_in[1];
  const float* w_out = (const float*)d_in[2];
  const float* b_out = (const float*)d_in[3];
  float* out = (float*)d_out;

  float* qkv   = (float*)d_ws;
  float* part  = qkv + 100663296L;
  float* ctxT  = part + 2097152L;
  float* kmaxb = ctxT + 131072L;
  float* kinvb = kmaxb + 2048L;

  const long N = NTOK;

  // 1) qkv[b] = w_qkv (768x128) @ x[b] (128xN)
  gemm_wmma_f32<128><<<dim3(24, 64, 8), 256, 0, stream>>>(
      w_qkv, x, qkv, nullptr,
      /*zdiv=*/1, 0, 0, 128 * N, 0, 768 * N, 0);

  // 2) k row softmax stats (max, 1/sumexp) per (b,h,d)
  k_rowstats<<<dim3(2048), 256, 0, stream>>>(qkv, kmaxb, kinvb);

  // 3) context^T partials over 16 n-chunks, then reduce
  context_wmma<<<dim3(8, 32), 256, 0, stream>>>(qkv, kmaxb, kinvb, part);
  ctx_reduce<<<dim3(512), 256, 0, stream>>>(part, ctxT);

  // 4) softmax q over d, in place
  q_softmax<<<dim3(2048), 256, 0, stream>>>(qkv);

  // 5) attn[b,h] (64xN) = ctxT[b,h] (64x64) @ qs[b,h] (64xN), into old k region
  gemm_wmma_f32<64><<<dim3(2, 64, 32), 256, 0, stream>>>(
      ctxT, qkv, qkv + 256 * N, nullptr,
      /*zdiv=*/4, 16384, 4096, 768 * N, 64 * N, 768 * N, 64 * N);

  // 6) out[b] = w_out (128x256) @ attn[b] (256xN) + b_out
  gemm_wmma_f32<256><<<dim3(4, 64, 8), 256, 0, stream>>>(
      w_out, qkv + 256 * N, out, b_out,
      /*zdiv=*/1, 0, 0, 768 * N, 0, 128 * N, 0);
}